// Sentence_49323404427376
// MI455X (gfx1250) — compile-verified
//
#include <hip/hip_runtime.h>

typedef __attribute__((ext_vector_type(2))) float v2f;
typedef __attribute__((ext_vector_type(4))) float v4f;
typedef __attribute__((ext_vector_type(8))) float v8f;

#define NB 512
#define MAXM 128
#define HD 1024
#define PC 128
#define TILE_M 64
#define KCH 128
#define NCH 8
#define LDSTX 132  // x staging stride: 128 + 4 pad (conflict-free, float4-aligned)
#define SST 136    // score tile stride: 128 + 8 pad

// ---------- packing ----------
__global__ void scan_kernel(const int* __restrict__ ms, int* __restrict__ offsets) {
  int acc = 0;
  for (int b = 0; b < NB; ++b) {
    offsets[b] = acc;
    int m = ms[b]; if (m > MAXM) m = MAXM;
    acc += m;
  }
  offsets[NB] = acc;
}

__global__ void fill_kernel(const int* __restrict__ ms, const int* __restrict__ offsets,
                            const int* __restrict__ labels,
                            int* __restrict__ row_b, int* __restrict__ row_p, int* __restrict__ yv) {
  int b = blockIdx.x;
  int m0 = ms[b]; if (m0 > MAXM) m0 = MAXM;
  int off = offsets[b];
  for (int m = threadIdx.x; m < m0; m += blockDim.x) {
    row_b[off + m] = b;
    row_p[off + m] = m;
    yv[off + m] = labels[b * MAXM + m];
  }
}

// ---------- small precomputes ----------
__global__ void pp_kernel(const float* __restrict__ proto, float* __restrict__ pp) {
  __shared__ float r[128];
  int p = blockIdx.x, j = threadIdx.x;
  float s = 0.f;
  for (int k = j; k < HD; k += 128) { float v = proto[(size_t)p * HD + k]; s += v * v; }
  r[j] = s; __syncthreads();
  if (j == 0) { float t = 0.f; for (int i = 0; i < 128; ++i) t += r[i]; pp[p] = t; }
}

__global__ void tvec_kernel(const float* __restrict__ vec_label,
                            const float* __restrict__ mat_label, float* __restrict__ tv) {
  int c = threadIdx.x;
  float s = 0.f;
  for (int p = 0; p < PC; ++p) {
    float z = mat_label[p * PC + c];
    s += vec_label[p] * (1.f / (1.f + expf(-z)));
  }
  tv[c] = s;
}

// ---------- fused main ----------
__global__ __launch_bounds__(256) void fused_main(
    const float* __restrict__ sent_embed, const float* __restrict__ pred_tok,
    const float* __restrict__ proto, const float* __restrict__ mloc,
    const float* __restrict__ vec_label, const float* __restrict__ mat_label,
    const int* __restrict__ row_b, const int* __restrict__ row_p,
    const int* __restrict__ yv, const float* __restrict__ pp,
    const float* __restrict__ tv, float* __restrict__ logits_out,
    float* __restrict__ partials, int N)
{
  // smem is a union: phase 1 = x staging (64 x KCH, stride 132 = 33.8 KB),
  //                  phase 2 = score/logits tile (64 x 128, stride 136 = 34.8 KB)
  __shared__ float smem[TILE_M * SST];
  __shared__ float red[256];
  __shared__ float rowtmp[TILE_M];
  __shared__ float xxsh[TILE_M];
  __shared__ int   ysh[TILE_M];
  __shared__ float accs[6];

  const int tid  = threadIdx.x;
  const int w    = tid >> 5;          // wave id 0..7
  const int lane = tid & 31;
  const int nl   = lane & 15;
  const int koff = (lane >> 4) << 1;  // 0 or 2
  const int moff = (lane >> 4) << 3;  // 0 or 8 (C-tile M offset)
  const int row  = tid >> 2;          // 0..63 (4 threads per row)
  const int t4   = tid & 3;
  const int n0   = blockIdx.x * TILE_M;

  // gather setup
  const int  nrow = n0 + row;
  const bool vr   = nrow < N;
  int rb = 0, rp = 0;
  if (vr) { rb = row_b[nrow]; rp = row_p[nrow]; }
  const v4f* s4 = (const v4f*)(sent_embed + ((size_t)rb * MAXM + rp) * HD);
  const v4f* p4 = (const v4f*)(pred_tok + (size_t)(vr ? nrow : 0) * HD);

  if (tid < TILE_M) ysh[tid] = (n0 + tid < N) ? yv[n0 + tid] : 0;

  const float* bp0  = proto + (size_t)(w * 16 + nl) * HD + koff;
  const float* bp1  = mloc  + (size_t)(w * 16 + nl) * HD + koff;
  const float* arow = smem + nl * LDSTX + koff;

  v8f cp0 = {}, cp1 = {}, cp2 = {}, cp3 = {};
  v8f cl0 = {}, cl1 = {}, cl2 = {}, cl3 = {};
  float xxacc = 0.f;

  for (int ch = 0; ch < NCH; ++ch) {
    __syncthreads();
    // stage K-chunk of 128 into LDS: per row, 32 float4s over 4 threads
    for (int j = 0; j < 8; ++j) {
      int c4 = j * 4 + t4;             // 0..31
      v4f v = {};
      if (vr) {
        int g = ch * 32 + c4;
        v4f a = s4[g]; v4f b = p4[g];
        v = a + b;
      }
      *(v4f*)&smem[row * LDSTX + c4 * 4] = v;
      xxacc += v[0]*v[0] + v[1]*v[1] + v[2]*v[2] + v[3]*v[3];
    }
    __syncthreads();
    const int kb = ch * KCH;
    for (int kk = 0; kk < KCH; kk += 4) {
      v2f b0 = *(const v2f*)(bp0 + kb + kk);
      v2f b1 = *(const v2f*)(bp1 + kb + kk);
      v2f a0 = *(const v2f*)(arow + 0  * LDSTX + kk);
      v2f a1 = *(const v2f*)(arow + 16 * LDSTX + kk);
      v2f a2 = *(const v2f*)(arow + 32 * LDSTX + kk);
      v2f a3 = *(const v2f*)(arow + 48 * LDSTX + kk);
      cp0 = __builtin_amdgcn_wmma_f32_16x16x4_f32(false, a0, false, b0, (short)0, cp0, false, false);
      cl0 = __builtin_amdgcn_wmma_f32_16x16x4_f32(false, a0, false, b1, (short)0, cl0, false, false);
      cp1 = __builtin_amdgcn_wmma_f32_16x16x4_f32(false, a1, false, b0, (short)0, cp1, false, false);
      cl1 = __builtin_amdgcn_wmma_f32_16x16x4_f32(false, a1, false, b1, (short)0, cl1, false, false);
      cp2 = __builtin_amdgcn_wmma_f32_16x16x4_f32(false, a2, false, b0, (short)0, cp2, false, false);
      cl2 = __builtin_amdgcn_wmma_f32_16x16x4_f32(false, a2, false, b1, (short)0, cl2, false, false);
      cp3 = __builtin_amdgcn_wmma_f32_16x16x4_f32(false, a3, false, b0, (short)0, cp3, false, false);
      cl3 = __builtin_amdgcn_wmma_f32_16x16x4_f32(false, a3, false, b1, (short)0, cl3, false, false);
    }
  }

  // barrier separates last A-fragment reads from reusing smem as the score tile
  red[tid] = xxacc;
  __syncthreads();

  float* sp = smem;  // now stride SST
  for (int r = 0; r < 8; ++r) {
    sp[(0  + r + moff) * SST + w * 16 + nl] = cp0[r];
    sp[(16 + r + moff) * SST + w * 16 + nl] = cp1[r];
    sp[(32 + r + moff) * SST + w * 16 + nl] = cp2[r];
    sp[(48 + r + moff) * SST + w * 16 + nl] = cp3[r];
  }
  if (tid < TILE_M) {
    float s = 0.f;
    for (int j = 0; j < 4; ++j) s += red[tid * 4 + j];
    xxsh[tid] = s;
  }
  __syncthreads();

  float* sprow = sp + row * SST;
  const float xxr = xxsh[row];
  const bool vrow = (n0 + row) < N;

  // scores = -max(0, ||x||^2 + ||p||^2 - 2 x.p - 1), then softmax over 128
  float lm = -1e30f;
  for (int i = 0; i < 32; ++i) {
    int p = t4 * 32 + i;
    float sc = -fmaxf(0.f, xxr + pp[p] - 2.f * sprow[p] - 1.f);
    sprow[p] = sc;
    lm = fmaxf(lm, sc);
  }
  red[tid] = lm; __syncthreads();
  if (t4 == 0) { float m = -1e30f; for (int j = 0; j < 4; ++j) m = fmaxf(m, red[row*4+j]); rowtmp[row] = m; }
  __syncthreads();
  const float rmax = rowtmp[row];
  float esum = 0.f;
  for (int i = 0; i < 32; ++i) {
    int p = t4 * 32 + i;
    float e = expf(sprow[p] - rmax);
    sprow[p] = e; esum += e;
  }
  red[tid] = esum; __syncthreads();
  if (t4 == 0) { float s = 0.f; for (int j = 0; j < 4; ++j) s += red[row*4+j]; rowtmp[row] = s; }
  __syncthreads();
  const float inv = 1.f / rowtmp[row];
  for (int i = 0; i < 32; ++i) {
    int p = t4 * 32 + i;
    float l = sprow[p] * inv;
    sprow[p] = l;
    if (vrow) logits_out[(size_t)(n0 + row) * PC + p] = l;
  }
  __syncthreads();

  // hinge + CE partials (per row, 4 threads x 32 cols)
  const int ly = ysh[row];
  const float correct = sprow[ly];
  float h = 0.f, ceexp = 0.f;
  for (int i = 0; i < 32; ++i) {
    int p = t4 * 32 + i;
    float l = sprow[p];
    ceexp += expf(l);
    if (p != ly) h += fmaxf(0.f, 1.f - correct + l);
  }
  if (!vrow) { h = 0.f; ceexp = 0.f; }

  red[tid] = h; __syncthreads();
  if (t4 == 0) { float s = 0.f; for (int j = 0; j < 4; ++j) s += red[row*4+j]; rowtmp[row] = s; }
  __syncthreads();
  if (tid == 0) { float s = 0.f; for (int m = 0; m < TILE_M; ++m) s += rowtmp[m]; accs[0] = s; }
  __syncthreads();

  red[tid] = ceexp; __syncthreads();
  if (t4 == 0) {
    float s = 0.f; for (int j = 0; j < 4; ++j) s += red[row*4+j];
    float v = 0.f;
    if (vrow) v = logf(s) - sp[row * SST + ly];
    rowtmp[row] = v;
  }
  __syncthreads();
  if (tid == 0) { float s = 0.f; for (int m = 0; m < TILE_M; ++m) s += rowtmp[m]; accs[1] = s; }
  __syncthreads();

  // local energies straight from the mat_local accumulators:
  // lane owns column p = w*16+nl of rows m = mt*16 + r + moff
  {
    const int pcol = w * 16 + nl;
    float elg = 0.f, ell = 0.f;
    for (int r = 0; r < 8; ++r) {
      int m0r = r + moff;
      if (n0 + m0r + 0  < N) { float l = sp[(m0r + 0 ) * SST + pcol]; elg += l * cl0[r]; if (ysh[m0r + 0 ] == pcol) ell += cl0[r]; }
      if (n0 + m0r + 16 < N) { float l = sp[(m0r + 16) * SST + pcol]; elg += l * cl1[r]; if (ysh[m0r + 16] == pcol) ell += cl1[r]; }
      if (n0 + m0r + 32 < N) { float l = sp[(m0r + 32) * SST + pcol]; elg += l * cl2[r]; if (ysh[m0r + 32] == pcol) ell += cl2[r]; }
      if (n0 + m0r + 48 < N) { float l = sp[(m0r + 48) * SST + pcol]; elg += l * cl3[r]; if (ysh[m0r + 48] == pcol) ell += cl3[r]; }
    }
    red[tid] = elg;
    __syncthreads();
    if (tid == 0) { float s = 0.f; for (int j = 0; j < 256; ++j) s += red[j]; accs[2] = s; }
    __syncthreads();
    red[tid] = ell;
    __syncthreads();
    if (tid == 0) { float s = 0.f; for (int j = 0; j < 256; ++j) s += red[j]; accs[3] = s; }
    __syncthreads();
  }

  // label energy (one-hot): sum tv[y]
  if (tid < TILE_M) red[tid] = (n0 + tid < N) ? tv[ysh[tid]] : 0.f;
  __syncthreads();
  if (tid == 0) { float s = 0.f; for (int m = 0; m < TILE_M; ++m) s += red[m]; accs[5] = s; }
  __syncthreads();

  // label energy (logits): out[m][p] = sum_c logits[m][c]*mat_label[p][c] via WMMA, K=128
  {
    v8f d0 = {}, d1 = {}, d2 = {}, d3 = {};
    const float* mrow  = mat_label + (size_t)(w * 16 + nl) * PC + koff;
    const float* arow2 = sp + nl * SST + koff;
    for (int kk = 0; kk < PC; kk += 4) {
      v2f bv = *(const v2f*)(mrow + kk);
      v2f a0 = *(const v2f*)(arow2 + 0  * SST + kk);
      v2f a1 = *(const v2f*)(arow2 + 16 * SST + kk);
      v2f a2 = *(const v2f*)(arow2 + 32 * SST + kk);
      v2f a3 = *(const v2f*)(arow2 + 48 * SST + kk);
      d0 = __builtin_amdgcn_wmma_f32_16x16x4_f32(false, a0, false, bv, (short)0, d0, false, false);
      d1 = __builtin_amdgcn_wmma_f32_16x16x4_f32(false, a1, false, bv, (short)0, d1, false, false);
      d2 = __builtin_amdgcn_wmma_f32_16x16x4_f32(false, a2, false, bv, (short)0, d2, false, false);
      d3 = __builtin_amdgcn_wmma_f32_16x16x4_f32(false, a3, false, bv, (short)0, d3, false, false);
    }
    const int p = w * 16 + nl;
    const float vl = vec_label[p];
    float s = 0.f;
    for (int r = 0; r < 8; ++r) {
      int m = r + moff;
      if (n0 + m + 0  < N) s += vl / (1.f + expf(-d0[r]));
      if (n0 + m + 16 < N) s += vl / (1.f + expf(-d1[r]));
      if (n0 + m + 32 < N) s += vl / (1.f + expf(-d2[r]));
      if (n0 + m + 48 < N) s += vl / (1.f + expf(-d3[r]));
    }
    red[tid] = s;
  }
  __syncthreads();
  if (tid == 0) { float s = 0.f; for (int j = 0; j < 256; ++j) s += red[j]; accs[4] = s; }
  __syncthreads();
  if (tid == 0) {
    float* pb = partials + (size_t)blockIdx.x * 6;
    for (int q = 0; q < 6; ++q) pb[q] = accs[q];
  }
}

// ---------- finalize ----------
__global__ void finalize_kernel(const float* __restrict__ partials, int nblk, int N,
                                float* __restrict__ out) {
  __shared__ float tot[6];
  int q = threadIdx.x;
  if (q < 6) {
    float s = 0.f;
    for (int b = 0; b < nblk; ++b) s += partials[(size_t)b * 6 + q];
    tot[q] = s;
  }
  __syncthreads();
  if (q == 0) {
    float loss_hinge  = tot[0] / ((float)N * (float)PC);
    float loss_ce     = tot[1] / (float)N;
    float E_label     = tot[3] + tot[5];
    float E_logits    = tot[2] + tot[4];
    float loss_energy = fmaxf(0.f, loss_hinge + E_label - E_logits);
    out[0] = 1.0f * loss_energy + 0.5f * loss_ce;
  }
}

__global__ void ycopy_kernel(const int* __restrict__ yv, float* __restrict__ out, int N) {
  int i = blockIdx.x * blockDim.x + threadIdx.x;
  if (i < N) out[i] = (float)yv[i];
}

extern "C" void kernel_launch(void* const* d_in, const int* in_sizes, int n_in,
                              void* d_out, int out_size, void* d_ws, size_t ws_size,
                              hipStream_t stream) {
  (void)n_in; (void)out_size; (void)ws_size;
  const float* sent_embed   = (const float*)d_in[0];
  const int*   sent_labels  = (const int*)d_in[1];
  const int*   mention_size = (const int*)d_in[2];
  const float* pred_tok     = (const float*)d_in[3];
  const float* proto        = (const float*)d_in[4];
  const float* mloc         = (const float*)d_in[5];
  const float* vec_label    = (const float*)d_in[6];
  const float* mat_label    = (const float*)d_in[7];
  float* out = (float*)d_out;

  const int N    = in_sizes[3] / HD;          // rows of pred_token_embedding
  const int nblk = (N + TILE_M - 1) / TILE_M;

  char* ws = (char*)d_ws;
  int*   offsets  = (int*)(ws);
  int*   row_b    = (int*)(ws + 4096);
  int*   row_p    = (int*)(ws + 4096 + 262144);
  int*   yv       = (int*)(ws + 4096 + 2 * 262144);
  float* pp       = (float*)(ws + 4096 + 3 * 262144);
  float* tv       = (float*)(ws + 4096 + 3 * 262144 + 512);
  float* partials = (float*)(ws + 4096 + 3 * 262144 + 1024);

  scan_kernel<<<1, 1, 0, stream>>>(mention_size, offsets);
  fill_kernel<<<NB, 128, 0, stream>>>(mention_size, offsets, sent_labels, row_b, row_p, yv);
  pp_kernel<<<PC, 128, 0, stream>>>(proto, pp);
  tvec_kernel<<<1, PC, 0, stream>>>(vec_label, mat_label, tv);
  fused_main<<<nblk, 256, 0, stream>>>(sent_embed, pred_tok, proto, mloc, vec_label, mat_label,
                                       row_b, row_p, yv, pp, tv, out + 1, partials, N);
  finalize_kernel<<<1, 32, 0, stream>>>(partials, nblk, N, out);
  ycopy_kernel<<<(N + 255) / 256, 256, 0, stream>>>(yv, out + 1 + (size_t)N * PC, N);
  hipMemcpyAsync(out + 1 + (size_t)N * PC + N, (const void*)proto,
                 (size_t)PC * HD * sizeof(float), hipMemcpyDeviceToDevice, stream);
}